// TaylorExp_14783277432863
// MI455X (gfx1250) — compile-verified
//
#include <hip/hip_runtime.h>

typedef __attribute__((ext_vector_type(2))) float v2f;
typedef __attribute__((ext_vector_type(8))) float v8f;

// Per position (B,H,S): out[0]=1, out[1..16]=x*0.5, out[17+i*16+j]=x_i*x_j*(1/(sqrt(2)*4))
// One wave32 handles one position; the 16x16 outer product is a single
// V_WMMA_F32_16X16X4_F32 with only the K=0 column/row populated.
__global__ void taylor_exp_wmma_kernel(const float* __restrict__ x,
                                       float* __restrict__ out,
                                       int npos) {
    const int lane = threadIdx.x & 31;
    const int wave = blockIdx.x * (blockDim.x >> 5) + (threadIdx.x >> 5);
    if (wave >= npos) return;   // wave-uniform branch: EXEC stays all-1s inside

    const float* xp = x + (size_t)wave * 16;
    // lanes 0-15 fetch x[0..15]; lanes 16-31 refetch the same (cheap, L0 hit)
    const float xv = __builtin_nontemporal_load(xp + (lane & 15));

    // A-matrix 16x4 f32: VGPR0 = K=0 (lanes 0-15) / K=2 (lanes 16-31); VGPR1 = K=1/K=3.
    // We want only K=0 nonzero. Scale A by 1/(sqrt(2)*sqrt(16)) so D = x2 directly.
    const float kInvS2 = 0.17677669529663687f;   // 1/(sqrt(2)*4)
    const bool lo = (lane < 16);
    v2f a; a.x = lo ? xv * kInvS2 : 0.0f; a.y = 0.0f;
    v2f b; b.x = lo ? xv          : 0.0f; b.y = 0.0f;
    v8f c = {};
    // D[m][n] = sum_k A[m][k] * B[k][n] = (x_m * kInvS2) * x_n
    v8f d = __builtin_amdgcn_wmma_f32_16x16x4_f32(
        /*neg_a=*/false, a, /*neg_b=*/false, b,
        /*c_mod=*/(short)0, c, /*reuse_a=*/false, /*reuse_b=*/false);

    float* op = out + (size_t)wave * 273;

    // term0 and term1 (RRD = sqrt(sqrt(16)) = 2 -> x * 0.5)
    if (lane == 0) __builtin_nontemporal_store(1.0f, op);
    if (lo)        __builtin_nontemporal_store(xv * 0.5f, op + 1 + lane);

    // x2: C/D layout: VGPR v, lanes 0-15 -> M=v, N=lane; lanes 16-31 -> M=v+8, N=lane-16
    const int row0 = (lane >> 4) << 3;  // 0 or 8
    const int col  = lane & 15;
    float* q = op + 17 + row0 * 16 + col;
#pragma unroll
    for (int v = 0; v < 8; ++v) {
        __builtin_nontemporal_store(d[v], q + v * 16);
    }
}

extern "C" void kernel_launch(void* const* d_in, const int* in_sizes, int n_in,
                              void* d_out, int out_size, void* d_ws, size_t ws_size,
                              hipStream_t stream) {
    const float* x = (const float*)d_in[0];
    float* out = (float*)d_out;
    const int npos = in_sizes[0] / 16;           // 4*16*4096 = 262144 positions
    const int wavesPerBlock = 8;                 // 256 threads = 8 wave32
    const int blocks = (npos + wavesPerBlock - 1) / wavesPerBlock;
    taylor_exp_wmma_kernel<<<blocks, 256, 0, stream>>>(x, out, npos);
}